// Ses2Seq_63127429316783
// MI455X (gfx1250) — compile-verified
//
#include <hip/hip_runtime.h>
#include <hip/hip_bf16.h>

// Problem constants (fixed by the reference)
#define LTOT   8448     // theta length
#define KTOT   65       // 1 + D
#define DDIM   64
#define HDIM   64
#define TSTEPS 128
#define BTOT   64
#define NB     8        // batches per workgroup
#define NTILES 528      // LTOT / 16
#define KCH    17       // ceil(65/4)  (K padded to 68)
#define TILES_PER_WAVE 66  // 528 / 8 waves

typedef float v2f __attribute__((ext_vector_type(2)));
typedef float v8f __attribute__((ext_vector_type(8)));

// ---------------- threefry2x32 (matches JAX's PRNG) ----------------
__device__ __forceinline__ unsigned rotl32(unsigned v, int r) {
    return (v << r) | (v >> (32 - r));
}
__device__ __forceinline__ void threefry2x32(unsigned k0, unsigned k1,
                                             unsigned x0, unsigned x1,
                                             unsigned& o0, unsigned& o1) {
    unsigned ks0 = k0, ks1 = k1, ks2 = 0x1BD11BDAu ^ k0 ^ k1;
    x0 += ks0; x1 += ks1;
    const int RA[4] = {13, 15, 26, 6};
    const int RB[4] = {17, 29, 16, 24};
    // 5 groups of 4 rounds with key injection
    #pragma unroll
    for (int g = 0; g < 5; ++g) {
        const int* R = (g & 1) ? RB : RA;
        #pragma unroll
        for (int i = 0; i < 4; ++i) { x0 += x1; x1 = rotl32(x1, R[i]); x1 ^= x0; }
        switch (g) {
            case 0: x0 += ks1; x1 += ks2 + 1u; break;
            case 1: x0 += ks2; x1 += ks0 + 2u; break;
            case 2: x0 += ks0; x1 += ks1 + 3u; break;
            case 3: x0 += ks1; x1 += ks2 + 4u; break;
            case 4: x0 += ks2; x1 += ks0 + 5u; break;
        }
    }
    o0 = x0; o1 = x1;
}

// ---------------- prep: retile Bmat into WMMA A-operand order ----------------
// tiledB[rt][kc][lane*2 + v]  where lane = m + 16*half, element (M=m, K=kc*4+2*half+v)
// K >= 65 is zero-padded.  One contiguous 8B load per lane per WMMA in the hot loop.
__global__ void prep_tiles_kernel(const float* __restrict__ Bmat,
                                  float* __restrict__ tiledB) {
    int idx = blockIdx.x * 256 + threadIdx.x;
    if (idx >= NTILES * KCH * 64) return;
    int v  = idx & 1;
    int l  = (idx >> 1) & 31;
    int kc = (idx >> 6) % KCH;
    int rt = idx / (64 * KCH);
    int m  = l & 15;
    int hl = l >> 4;
    int k  = kc * 4 + 2 * hl + v;
    int row = rt * 16 + m;
    tiledB[idx] = (k < KTOT) ? Bmat[row * KTOT + k] : 0.0f;
}

// ---------------- main scan kernel: 8 blocks x 256 threads (8 waves) ----------------
// LDS layout (floats):
//   theta : 8448*8              (f32, per-batch theta columns, resident all 128 steps)
//   dz    : 68*16               (K padded to 68, N padded to 16; pad entries stay 0)
//   zbuf  : 8*64,  zbuf2 : 8*64, ybuf : 8*64
//   forced: 8*128               (precomputed bernoulli bits as 0/1 floats)
#define TH_F   (LTOT * NB)          // 67584
#define DZ_F   (68 * 16)            // 1088
#define ZB_F   (NB * HDIM)          // 512
#define SMEM_FLOATS (TH_F + DZ_F + 3 * ZB_F + NB * TSTEPS)

__global__ void ses2seq_scan_kernel(const float* __restrict__ xs,
                                    const float* __restrict__ theta0,
                                    const unsigned* __restrict__ key,
                                    const float* __restrict__ tiledB,
                                    float* __restrict__ out) {
    extern __shared__ float smem[];
    float* theta  = smem;
    float* dzs    = theta + TH_F;
    float* zbuf   = dzs + DZ_F;
    float* zbuf2  = zbuf + ZB_F;
    float* ybuf   = zbuf2 + ZB_F;
    float* forced = ybuf + ZB_F;

    const int tid  = threadIdx.x;
    const int wave = tid >> 5;          // 0..7 : local batch for DZ/MLP phases
    const int lane = tid & 31;
    const int gb0  = blockIdx.x * NB;   // first global batch of this workgroup

    // ---- init theta (broadcast theta0 into 8 batch columns) ----
    for (int i = tid; i < LTOT; i += 256) {
        float v = theta0[i];
        #pragma unroll
        for (int c = 0; c < NB; ++c) theta[i * NB + c] = v;
    }
    // ---- init dz padding (cols 8..15 and k rows 65..67 must stay zero) ----
    for (int i = tid; i < DZ_F; i += 256) dzs[i] = 0.0f;
    // ---- precompute teacher-forcing bits: split(key,64)->split(.,1)->split(.,128)->bernoulli ----
    const unsigned k0 = key[0], k1 = key[1];
    for (int idx = tid; idx < NB * TSTEPS; idx += 256) {
        int b = idx >> 7, t = idx & 127;
        unsigned kb0, kb1, lk0, lk1, sk0, sk1, r0, r1;
        threefry2x32(k0, k1, (unsigned)(gb0 + b), (unsigned)(BTOT + gb0 + b), kb0, kb1);
        threefry2x32(kb0, kb1, 0u, 1u, lk0, lk1);
        threefry2x32(lk0, lk1, (unsigned)t, (unsigned)(TSTEPS + t), sk0, sk1);
        threefry2x32(sk0, sk1, 0u, 0u, r0, r1);
        float u = __uint_as_float((r0 >> 9) | 0x3f800000u) - 1.0f;
        forced[idx] = (u < 0.9f) ? 1.0f : 0.0f;
    }
    __syncthreads();

    const int n  = lane & 15;     // WMMA N index (batch column; valid if < 8)
    const int hl = lane >> 4;     // WMMA half selector
    const int nc = n & 7;         // clamped column for C-operand loads

    for (int t = 0; t < TSTEPS; ++t) {
        // ================= phase A: build DZ columns (wave b handles batch b) =================
        {
            const int b  = wave;
            const int gb = gb0 + b;
            if (lane == 0) dzs[0 * 16 + b] = 1.0f / (float)TSTEPS;  // dt is constant
            const float f = forced[b * TSTEPS + t];
            #pragma unroll
            for (int rep = 0; rep < 2; ++rep) {
                int i = lane + 32 * rep;
                float d = 0.0f;
                if (t > 0 && f > 0.5f) {
                    float xt = xs[((size_t)gb * TSTEPS + t) * DDIM + i];
                    float yp = ybuf[b * DDIM + i];
                    d = xt - yp;
                }
                dzs[(1 + i) * 16 + b] = d;
            }
        }
        __syncthreads();

        // ================= phase B: theta += Bmat @ DZ  via V_WMMA_F32_16X16X4_F32 =============
        {
            // B-operand (DZ) loaded once per step, reused across all 66 tiles of this wave.
            v2f Bv[KCH];
            #pragma unroll
            for (int kc = 0; kc < KCH; ++kc) {
                int k = kc * 4 + 2 * hl;
                Bv[kc].x = dzs[k * 16 + n];
                Bv[kc].y = dzs[(k + 1) * 16 + n];
            }
            const int rt0 = wave * TILES_PER_WAVE;
            for (int rt = rt0; rt < rt0 + TILES_PER_WAVE; ++rt) {
                const int rowbase = rt * 16 + 8 * hl;     // C tile: lane covers rows rowbase..+7, col n
                v8f c;
                #pragma unroll
                for (int r = 0; r < 8; ++r) c[r] = theta[(rowbase + r) * NB + nc];
                const float* ap = tiledB + (size_t)rt * (KCH * 64) + lane * 2;
                __builtin_prefetch(ap + KCH * 64, 0, 3);  // next tile -> global_prefetch
                #pragma unroll
                for (int kc = 0; kc < KCH; ++kc) {
                    v2f a = *(const v2f*)(ap + kc * 64);
                    c = __builtin_amdgcn_wmma_f32_16x16x4_f32(
                            /*neg_a=*/false, a, /*neg_b=*/false, Bv[kc],
                            /*c_mod=*/(short)0, c, /*reuse_a=*/false, /*reuse_b=*/false);
                }
                if (n < NB) {   // store only the 8 real batch columns
                    #pragma unroll
                    for (int r = 0; r < 8; ++r) theta[(rowbase + r) * NB + n] = c[r];
                }
            }
        }
        __syncthreads();

        // ================= phase C: per-batch MLP (wave b = batch b) ===========================
        {
            const int b  = wave;
            const float txy = (float)(t + 1) * (1.0f / (float)TSTEPS);  // 2*t_curr - t_prev
            #pragma unroll
            for (int rep = 0; rep < 2; ++rep) {
                int h = lane + 32 * rep;
                float w0 = theta[h * NB + b];
                float b0 = theta[(64 + h) * NB + b];
                zbuf[b * HDIM + h] = fmaxf(w0 * txy + b0, 0.0f);
            }
            __syncthreads();
            #pragma unroll
            for (int rep = 0; rep < 2; ++rep) {
                int h = lane + 32 * rep;
                float acc = theta[(4224 + h) * NB + b];          // b1
                for (int j = 0; j < HDIM; ++j) {
                    int jj = (j + lane) & 63;                    // skew to spread LDS banks
                    acc += theta[(128 + h * 64 + jj) * NB + b] * zbuf[b * HDIM + jj];
                }
                zbuf2[b * HDIM + h] = fmaxf(acc, 0.0f);
            }
            __syncthreads();
            const int gb = gb0 + b;
            #pragma unroll
            for (int rep = 0; rep < 2; ++rep) {
                int d = lane + 32 * rep;
                float acc = theta[(8384 + d) * NB + b];          // b2
                for (int j = 0; j < HDIM; ++j) {
                    int jj = (j + lane) & 63;
                    acc += theta[(4288 + d * 64 + jj) * NB + b] * zbuf2[b * HDIM + jj];
                }
                float x = fminf(fmaxf(acc, -15.0f), 15.0f);
                float e = __expf(2.0f * x);
                float y = (e - 1.0f) / (e + 1.0f);               // tanh
                ybuf[b * DDIM + d] = y;
                out[((size_t)gb * TSTEPS + t) * DDIM + d] = y;
            }
        }
        __syncthreads();
    }
}

extern "C" void kernel_launch(void* const* d_in, const int* in_sizes, int n_in,
                              void* d_out, int out_size, void* d_ws, size_t ws_size,
                              hipStream_t stream) {
    const float*    xs     = (const float*)d_in[0];
    // d_in[1] = ts (unused: dt == 1/T exactly), d_in[2] = A (identity, unused)
    const float*    Bmat   = (const float*)d_in[3];
    const float*    theta0 = (const float*)d_in[4];
    const unsigned* key    = (const unsigned*)d_in[5];
    float*          out    = (float*)d_out;
    float*          tiledB = (float*)d_ws;   // 528*17*64 floats = 2.3 MB, lives in L2

    const int prep_elems = NTILES * KCH * 64;
    prep_tiles_kernel<<<(prep_elems + 255) / 256, 256, 0, stream>>>(Bmat, tiledB);

    const size_t smem_bytes = (size_t)SMEM_FLOATS * sizeof(float);  // ~285 KB of the 320 KB WGP LDS
    ses2seq_scan_kernel<<<BTOT / NB, 256, smem_bytes, stream>>>(xs, theta0, key, tiledB, out);
}